// PointEncoderV6_87995289960541
// MI455X (gfx1250) — compile-verified
//
#include <hip/hip_runtime.h>
#include <math.h>

// ---------------------------------------------------------------------------
// CDNA5 fp32 WMMA helpers: D(16x16) = A(16x4) * B(4x16) + C, wave32.
// A layout: lane l holds row m=l&15, k = 2*(l>>4)+{0,1} in the 2 VGPRs.
// B layout: lane l holds col n=l&15, k = 2*(l>>4)+{0,1}.
// C/D layout: VGPR r: row = r + 8*(l>>4), col = l&15.
// ---------------------------------------------------------------------------
typedef __attribute__((ext_vector_type(2))) float v2f;
typedef __attribute__((ext_vector_type(8))) float v8f;

__device__ __forceinline__ v8f wmma4(v2f a, v2f b, v8f c) {
  return __builtin_amdgcn_wmma_f32_16x16x4_f32(false, a, false, b, (short)0, c,
                                               false, false);
}

#define CDIM 256

// ---------------------------------------------------------------------------
// Generic Y = post(X @ W + bias), X:(M,256) W:(256,256).
// mode 0: Y = X@W + bias          mode 1: Y = LN(relu(X@W + bias), gamma, beta)
// block: 256 thr (8 waves) -> 16 rows x 256 cols tile; each wave: 2 n-tiles.
// ---------------------------------------------------------------------------
__global__ __launch_bounds__(256) void gemm256_kernel(
    const float* __restrict__ X, const float* __restrict__ W,
    const float* __restrict__ bias, const float* __restrict__ gamma,
    const float* __restrict__ beta, float* __restrict__ Y, int M, int mode) {
  __shared__ float Xs[16][CDIM];
  __shared__ float Ys[16][CDIM];
  __shared__ float mu[16], rstd[16];
  const int row0 = blockIdx.x * 16;
  const int tid = threadIdx.x;
  const int lane = tid & 31;
  const int wave = tid >> 5;

  for (int e = tid; e < 16 * CDIM; e += 256) {
    int r = e >> 8, c = e & 255;
    Xs[r][c] = X[(size_t)(row0 + r) * CDIM + c];
  }
  __syncthreads();

  const int m = lane & 15;
  const int kk = (lane >> 4) * 2;
  const int mbase = (lane >> 4) * 8;

  for (int t = 0; t < 2; ++t) {
    const int n = (wave * 2 + t) * 16 + (lane & 15);
    v8f acc = {};
    for (int k0 = 0; k0 < CDIM; k0 += 4) {
      v2f a, b;
      a.x = Xs[m][k0 + kk];
      a.y = Xs[m][k0 + kk + 1];
      b.x = W[(size_t)(k0 + kk) * CDIM + n];
      b.y = W[(size_t)(k0 + kk + 1) * CDIM + n];
      acc = wmma4(a, b, acc);
    }
    const float bv = bias ? bias[n] : 0.f;
#pragma unroll
    for (int r = 0; r < 8; ++r) {
      float v = acc[r] + bv;
      if (mode == 1) v = fmaxf(v, 0.f);
      Ys[mbase + r][n] = v;
    }
  }
  __syncthreads();

  if (mode == 1) {
    if (tid < 16) {
      float s = 0.f;
      for (int c = 0; c < CDIM; ++c) s += Ys[tid][c];
      const float mean = s * (1.f / CDIM);
      float v = 0.f;
      for (int c = 0; c < CDIM; ++c) {
        float d = Ys[tid][c] - mean;
        v += d * d;
      }
      mu[tid] = mean;
      rstd[tid] = rsqrtf(v * (1.f / CDIM) + 1e-5f);
    }
    __syncthreads();
    for (int e = tid; e < 16 * CDIM; e += 256) {
      int r = e >> 8, c = e & 255;
      Y[(size_t)(row0 + r) * CDIM + c] =
          (Ys[r][c] - mu[r]) * rstd[r] * gamma[c] + beta[c];
    }
  } else {
    for (int e = tid; e < 16 * CDIM; e += 256) {
      int r = e >> 8, c = e & 255;
      Y[(size_t)(row0 + r) * CDIM + c] = Ys[r][c];
    }
  }
}

// roi = sigmoid(x @ w3 + b3), w3:(256,1)
__global__ void roi_kernel(const float* __restrict__ x,
                           const float* __restrict__ w3,
                           const float* __restrict__ b3,
                           float* __restrict__ roi, int M) {
  int i = blockIdx.x * blockDim.x + threadIdx.x;
  if (i >= M) return;
  float s = 0.f;
  for (int c = 0; c < CDIM; ++c) s += x[(size_t)i * CDIM + c] * w3[c];
  s += b3[0];
  roi[i] = 1.f / (1.f + expf(-s));
}

// Per-agent top-256 of 1024 via bit-pattern binary search (positive floats ->
// monotone uint bits). Ties taken at lowest index (matches stable top_k set).
__global__ __launch_bounds__(256) void topk_kernel(const float* __restrict__ roi,
                                                   int* __restrict__ top_idx,
                                                   float* __restrict__ conf) {
  __shared__ unsigned bits[1024];
  __shared__ int red;
  const int n = blockIdx.x;
  const int tid = threadIdx.x;
  for (int e = tid; e < 1024; e += 256)
    bits[e] = __float_as_uint(roi[n * 1024 + e]);
  __syncthreads();
  unsigned lo = 0u, hi = 0xFFFFFFFFu;
  while (lo < hi) {
    unsigned mid = lo + ((hi - lo) >> 1) + 1u;
    if (tid == 0) red = 0;
    __syncthreads();
    int c = 0;
    for (int e = tid; e < 1024; e += 256)
      if (bits[e] >= mid) ++c;
    atomicAdd(&red, c);
    __syncthreads();
    int total = red;
    __syncthreads();
    if (total >= 256) lo = mid;
    else hi = mid - 1u;
  }
  if (tid == 0) {
    int c = 0;
    for (int e = 0; e < 1024 && c < 256; ++e) {
      if (bits[e] > lo) {
        top_idx[n * 256 + c] = e;
        conf[n * 256 + c] = (__uint_as_float(bits[e]) > 0.7f) ? 1.f : 0.f;
        ++c;
      }
    }
    for (int e = 0; e < 1024 && c < 256; ++e) {
      if (bits[e] == lo) {
        top_idx[n * 256 + c] = e;
        conf[n * 256 + c] = (__uint_as_float(bits[e]) > 0.7f) ? 1.f : 0.f;
        ++c;
      }
    }
  }
}

// nf_masked[n,k,c] = b_feature[n, top_idx[n,k], c] * conf[n,k]
__global__ void gather_nf_kernel(const float* __restrict__ bf,
                                 const int* __restrict__ idx,
                                 const float* __restrict__ conf,
                                 float* __restrict__ nf) {
  int g = blockIdx.x * 256 + threadIdx.x;  // 6*256*256 exact
  int c = g & 255;
  int k = (g >> 8) & 255;
  int n = g >> 16;
  nf[g] = bf[((size_t)n * 1024 + idx[n * 256 + k]) * CDIM + c] *
          conf[n * 256 + k];
}

// pts[i,m,d] = i2j_anchor[i, j(m), top_idx[j(m), m&255], d], d<3, m in [0,1280)
__global__ void gather_pts_kernel(const float* __restrict__ i2j,
                                  const int* __restrict__ idx,
                                  float* __restrict__ pts, int total) {
  int g = blockIdx.x * blockDim.x + threadIdx.x;
  if (g >= total) return;
  int d = g % 3;
  int m = (g / 3) % 1280;
  int i = g / (3 * 1280);
  int jj = m >> 8;
  int j = jj + (jj >= i);
  int a = idx[j * 256 + (m & 255)];
  pts[g] = i2j[(((size_t)i * 6 + j) * 1024 + a) * 8 + d];
}

__global__ void ew_add_kernel(const float* a, const float* b, float* y, int n) {
  int e = blockIdx.x * blockDim.x + threadIdx.x;
  if (e < n) y[e] = a[e] + b[e];
}
__global__ void ew_sum5_kernel(const float* o, float* y, int n) {
  int e = blockIdx.x * blockDim.x + threadIdx.x;
  if (e < n)
    y[e] = o[e] + o[n + e] + o[2 * (size_t)n + e] + o[3 * (size_t)n + e] +
           o[4 * (size_t)n + e];
}
// res = rtmp + 4*bo + 5*ego   (gemm already added bo once; sum over 5 batches)
__global__ void ew_res_kernel(const float* rtmp, const float* bo,
                              const float* ego, float* y, int n) {
  int e = blockIdx.x * blockDim.x + threadIdx.x;
  if (e < n) y[e] = rtmp[e] + 4.f * bo[e & 255] + 5.f * ego[e];
}
__global__ void copy_kernel(const float* a, float* y, int n) {
  int e = blockIdx.x * blockDim.x + threadIdx.x;
  if (e < n) y[e] = a[e];
}

// ---------------------------------------------------------------------------
// Attention: one block per (qtile, head, batch). 4 waves.
// S = softmax(Q Kt / sqrt(32)) fully materialized (16x1024 in LDS), then AV.
// ---------------------------------------------------------------------------
__global__ __launch_bounds__(128) void attn_kernel(
    const float* __restrict__ q, const float* __restrict__ kmat,
    const float* __restrict__ vmat, float* __restrict__ o_all) {
  __shared__ float S[16][1024];  // 64 KB
  __shared__ float Qs[16][32];
  __shared__ float rinv[16];
  const int qt = blockIdx.x, h = blockIdx.y, b = blockIdx.z;
  const int tid = threadIdx.x;
  const int lane = tid & 31;
  const int wave = tid >> 5;  // 0..3
  const int q0 = qt * 16;
  const float* Qp = q + (size_t)q0 * CDIM + h * 32;
  const float* Kp = kmat + (size_t)b * 1024 * CDIM + h * 32;
  const float* Vp = vmat + (size_t)b * 1024 * CDIM + h * 32;

  for (int e = tid; e < 16 * 32; e += 128) {
    int r = e >> 5, c = e & 31;
    Qs[r][c] = Qp[(size_t)r * CDIM + c];
  }
  __syncthreads();

  const int m = lane & 15;
  const int kk = (lane >> 4) * 2;
  const int mbase = (lane >> 4) * 8;
  const float scale = 0.1767766952966369f;  // 1/sqrt(32)

  for (int kt = wave; kt < 64; kt += 4) {
    const int key0 = kt * 16;
    const int n = key0 + (lane & 15);
    v8f acc = {};
    for (int k0 = 0; k0 < 32; k0 += 4) {
      v2f a, bb;
      a.x = Qs[m][k0 + kk];
      a.y = Qs[m][k0 + kk + 1];
      bb.x = Kp[(size_t)n * CDIM + k0 + kk];      // B[k][n] = K[n][k]
      bb.y = Kp[(size_t)n * CDIM + k0 + kk + 1];
      acc = wmma4(a, bb, acc);
    }
#pragma unroll
    for (int r = 0; r < 8; ++r) S[mbase + r][n] = acc[r] * scale;
  }
  __syncthreads();

  if (tid < 16) {
    float mx = -1e30f;
    for (int e = 0; e < 1024; ++e) mx = fmaxf(mx, S[tid][e]);
    float s = 0.f;
    for (int e = 0; e < 1024; ++e) {
      float ex = expf(S[tid][e] - mx);
      S[tid][e] = ex;
      s += ex;
    }
    rinv[tid] = 1.f / s;
  }
  __syncthreads();
  for (int e = tid; e < 16 * 1024; e += 128) {
    int r = e >> 10, c = e & 1023;
    S[r][c] *= rinv[r];
  }
  __syncthreads();

  if (wave < 2) {  // wave-uniform: EXEC full inside each executing wave
    const int n = wave * 16 + (lane & 15);
    v8f acc = {};
    for (int k0 = 0; k0 < 1024; k0 += 4) {
      v2f a, bb;
      a.x = S[m][k0 + kk];
      a.y = S[m][k0 + kk + 1];
      bb.x = Vp[(size_t)(k0 + kk) * CDIM + n];
      bb.y = Vp[(size_t)(k0 + kk + 1) * CDIM + n];
      acc = wmma4(a, bb, acc);
    }
    float* Op = o_all + (size_t)b * 1024 * CDIM + (size_t)q0 * CDIM + h * 32;
#pragma unroll
    for (int r = 0; r < 8; ++r) Op[(size_t)(mbase + r) * CDIM + n] = acc[r];
  }
}

// ---------------------------------------------------------------------------
// out_feat[i,a,c] = res[a,c] + (inside(i,a,:) . nf_rm(i,:,c)) / max(cnt,1)
// inside generated on the fly from box tests; (1024x1280)x(1280x256) per agent.
// ---------------------------------------------------------------------------
__global__ __launch_bounds__(256) void agg_kernel(
    const float* __restrict__ b_anchor, const float* __restrict__ pts,
    const float* __restrict__ nf, const float* __restrict__ res,
    float* __restrict__ out) {
  __shared__ float mn[16][3], mx[16][3];
  __shared__ float cbuf[32];
  __shared__ float cnt_sh[16];
  const int i = blockIdx.y;
  const int a0 = blockIdx.x * 16;
  const int tid = threadIdx.x;
  const int lane = tid & 31;
  const int wave = tid >> 5;

  if (tid < 16) {
    const float* ba = b_anchor + ((size_t)i * 1024 + a0 + tid) * 8;
    float s = fabsf(ba[6]);
    float c = fabsf(ba[7]);
    float d0 = expf(ba[3]), d1 = expf(ba[4]), d2 = expf(ba[5]);
    float hx = 0.5f * (d0 * c + d1 * s);
    float hy = 0.5f * (d0 * s + d1 * c);
    float hz = 0.5f * d2;
    mn[tid][0] = ba[0] - hx; mx[tid][0] = ba[0] + hx;
    mn[tid][1] = ba[1] - hy; mx[tid][1] = ba[1] + hy;
    mn[tid][2] = ba[2] - hz; mx[tid][2] = ba[2] + hz;
  }
  __syncthreads();

  const int m = lane & 15;
  const int kk = (lane >> 4) * 2;
  const int mbase = (lane >> 4) * 8;
  const int n0 = wave * 32 + (lane & 15);
  const int n1 = n0 + 16;
  const float* Pi = pts + (size_t)i * 1280 * 3;

  v8f acc0 = {}, acc1 = {};
  float cpart = 0.f;  // identical in every wave (wave 0 publishes)
  for (int k0 = 0; k0 < 1280; k0 += 4) {
    const int mm = k0 + kk;  // even, mm and mm+1 share the same 256-block
    const float* p = Pi + (size_t)mm * 3;
    v2f a;
    a.x = (p[0] >= mn[m][0] && p[0] <= mx[m][0] && p[1] >= mn[m][1] &&
           p[1] <= mx[m][1] && p[2] >= mn[m][2] && p[2] <= mx[m][2])
              ? 1.f : 0.f;
    a.y = (p[3] >= mn[m][0] && p[3] <= mx[m][0] && p[4] >= mn[m][1] &&
           p[4] <= mx[m][1] && p[5] >= mn[m][2] && p[5] <= mx[m][2])
              ? 1.f : 0.f;
    cpart += a.x + a.y;
    int jj = mm >> 8;
    int j = jj + (jj >= i);
    const float* nfb = nf + ((size_t)j * 256 + (mm & 255)) * CDIM;
    v2f b0, b1;
    b0.x = nfb[n0];
    b0.y = nfb[CDIM + n0];
    b1.x = nfb[n1];
    b1.y = nfb[CDIM + n1];
    acc0 = wmma4(a, b0, acc0);
    acc1 = wmma4(a, b1, acc1);
  }
  if (wave == 0) cbuf[lane] = cpart;
  __syncthreads();
  if (tid < 16) cnt_sh[tid] = cbuf[tid] + cbuf[tid + 16];
  __syncthreads();

  float* Op = out + ((size_t)i * 1024 + a0) * CDIM;
  const float* Rp = res + (size_t)a0 * CDIM;
#pragma unroll
  for (int r = 0; r < 8; ++r) {
    const int row = mbase + r;
    const float div = 1.f / fmaxf(cnt_sh[row], 1.f);
    Op[(size_t)row * CDIM + n0] = Rp[(size_t)row * CDIM + n0] + acc0[r] * div;
    Op[(size_t)row * CDIM + n1] = Rp[(size_t)row * CDIM + n1] + acc1[r] * div;
  }
}

// ---------------------------------------------------------------------------
extern "C" void kernel_launch(void* const* d_in, const int* in_sizes, int n_in,
                              void* d_out, int out_size, void* d_ws,
                              size_t ws_size, hipStream_t stream) {
  (void)in_sizes; (void)n_in; (void)out_size; (void)ws_size;
  const float* i2j      = (const float*)d_in[0];
  const float* b_anchor = (const float*)d_in[1];
  const float* b_feat   = (const float*)d_in[2];
  const float* a_embed  = (const float*)d_in[5];
  const float* cls_w1 = (const float*)d_in[6];
  const float* cls_b1 = (const float*)d_in[7];
  const float* cls_g1 = (const float*)d_in[8];
  const float* cls_bb1 = (const float*)d_in[9];
  const float* cls_w2 = (const float*)d_in[10];
  const float* cls_b2 = (const float*)d_in[11];
  const float* cls_g2 = (const float*)d_in[12];
  const float* cls_bb2 = (const float*)d_in[13];
  const float* cls_w3 = (const float*)d_in[14];
  const float* cls_b3 = (const float*)d_in[15];
  const float* wq = (const float*)d_in[26];
  const float* bq = (const float*)d_in[27];
  const float* wk = (const float*)d_in[28];
  const float* bk = (const float*)d_in[29];
  const float* wv = (const float*)d_in[30];
  const float* bv = (const float*)d_in[31];
  const float* wo = (const float*)d_in[32];
  const float* bo = (const float*)d_in[33];

  float* ws = (float*)d_ws;
  // bump layout (floats); X1/X2 recycled as K/V after roi is consumed.
  float* X1   = ws + 0;        // 6144*256
  float* X2   = ws + 1572864;  // 6144*256
  float* ROI  = ws + 3145728;  // 6144
  int*   IDX  = (int*)(ws + 3151872);  // 6*256
  float* CONF = ws + 3153408;  // 6*256
  float* NF   = ws + 3154944;  // 6*256*256
  float* PTS  = ws + 3548160;  // 6*1280*3
  float* QIN  = ws + 3571200;  // 1024*256
  float* Q    = ws + 3833344;  // 1024*256
  float* OALL = ws + 4095488;  // 5*1024*256
  float* SO   = ws + 5406208;  // 1024*256
  float* RTMP = ws + 5668352;  // 1024*256
  float* RES  = ws + 5930496;  // 1024*256   (total ~24.8 MB)
  float* KK = X1;
  float* VV = X2;

  // 1) cls MLP -> roi
  gemm256_kernel<<<384, 256, 0, stream>>>(b_feat, cls_w1, cls_b1, cls_g1,
                                          cls_bb1, X1, 6144, 1);
  gemm256_kernel<<<384, 256, 0, stream>>>(X1, cls_w2, cls_b2, cls_g2, cls_bb2,
                                          X2, 6144, 1);
  roi_kernel<<<24, 256, 0, stream>>>(X2, cls_w3, cls_b3, ROI, 6144);

  // 2) top-256 per agent + conf mask
  topk_kernel<<<6, 256, 0, stream>>>(ROI, IDX, CONF);

  // 3) gathers
  gather_nf_kernel<<<1536, 256, 0, stream>>>(b_feat, IDX, CONF, NF);
  gather_pts_kernel<<<90, 256, 0, stream>>>(i2j, IDX, PTS, 6 * 1280 * 3);

  // 4) attention
  ew_add_kernel<<<1024, 256, 0, stream>>>(b_feat, a_embed, QIN, 1024 * 256);
  gemm256_kernel<<<64, 256, 0, stream>>>(QIN, wq, bq, nullptr, nullptr, Q,
                                         1024, 0);
  gemm256_kernel<<<320, 256, 0, stream>>>(b_feat + 1024 * 256, wk, bk, nullptr,
                                          nullptr, KK, 5120, 0);
  gemm256_kernel<<<320, 256, 0, stream>>>(b_feat + 1024 * 256, wv, bv, nullptr,
                                          nullptr, VV, 5120, 0);
  dim3 ag(64, 8, 5);
  attn_kernel<<<ag, 128, 0, stream>>>(Q, KK, VV, OALL);
  ew_sum5_kernel<<<1024, 256, 0, stream>>>(OALL, SO, 1024 * 256);
  gemm256_kernel<<<64, 256, 0, stream>>>(SO, wo, bo, nullptr, nullptr, RTMP,
                                         1024, 0);
  ew_res_kernel<<<1024, 256, 0, stream>>>(RTMP, bo, b_feat, RES, 1024 * 256);

  // 5) outputs: b_anchor passthrough, then fused inside-GEMM + res epilogue
  copy_kernel<<<192, 256, 0, stream>>>(b_anchor, (float*)d_out, 6 * 1024 * 8);
  dim3 gg(64, 6);
  agg_kernel<<<gg, 256, 0, stream>>>(b_anchor, PTS, NF, RES,
                                     (float*)d_out + 6 * 1024 * 8);
}